// LearnedRouter_88089779241156
// MI455X (gfx1250) — compile-verified
//
#include <hip/hip_runtime.h>
#include <hip/hip_bf16.h>
#include <stdint.h>

// Problem sizes: hidden [4,8192,2048] f32, gate_w [64,2048] f32, gate_b [64] f32.
// Outputs: weights f32 [4,8192,2] then topk_idx i32 [4,8192,2] (concatenated in d_out).
#define H       2048
#define E       64
#define MT      128     // tokens per workgroup
#define KC      64      // K-chunk staged in LDS
#define T_TOTAL 32768   // 4*8192 tokens

typedef __attribute__((ext_vector_type(2))) float v2f;
typedef __attribute__((ext_vector_type(8))) float v8f;

#define AS_STRIDE 68    // floats per A row (16B aligned rows, conflict-free b64 frags)
#define BS_STRIDE 68    // floats per B row (same properties)
#define L_STRIDE  65    // floats per logits row (conflict-free serial scan)

// ---- CDNA5 async global->LDS (ASYNCcnt-tracked, bypasses VGPRs) ----
__device__ __forceinline__ void async_load_b128_to_lds(uint32_t lds_addr, const float* gptr) {
    asm volatile("global_load_async_to_lds_b128 %0, %1, off"
                 :: "v"(lds_addr), "v"((uint64_t)(uintptr_t)gptr)
                 : "memory");
}

#if __has_builtin(__builtin_amdgcn_s_wait_asynccnt)
#define WAIT_ASYNCCNT(n) __builtin_amdgcn_s_wait_asynccnt(n)
#else
#define WAIT_ASYNCCNT(n) asm volatile("s_wait_asynccnt %0" :: "i"(n) : "memory")
#endif

__device__ __forceinline__ v2f ld2(const float* p) {
    const float2 t = *reinterpret_cast<const float2*>(p);
    v2f r; r.x = t.x; r.y = t.y; return r;
}

__global__ __launch_bounds__(256) void LearnedRouter_kernel(
    const float* __restrict__ hidden,
    const float* __restrict__ gate_w,
    const float* __restrict__ gate_b,
    float* __restrict__ out_w,
    int*   __restrict__ out_i)
{
    __shared__ float As[2][MT * AS_STRIDE];   // 2 x 34816 B (buf 0 reused for logits)
    __shared__ float Bs[2][E * BS_STRIDE];    // 2 x 17408 B
    __shared__ float bias_s[E];

    const int tid  = threadIdx.x;
    const int wave = tid >> 5;
    const int lane = tid & 31;
    const int n    = lane & 15;   // column / row-in-stripe
    const int hi   = lane >> 4;   // half-wave K-pair select
    const int m0   = blockIdx.x * MT;

    if (tid < E) bias_s[tid] = gate_b[tid];

    const uint32_t as_base = (uint32_t)(uintptr_t)&As[0][0];
    const uint32_t bs_base = (uint32_t)(uintptr_t)&Bs[0][0];
    const uint32_t abuf    = (uint32_t)(MT * AS_STRIDE * 4);
    const uint32_t bbuf    = (uint32_t)(E * BS_STRIDE * 4);

    // Issue one K-chunk's DMA: 8 (A) + 4 (B) async b128 per thread = 12 ops
    auto issue = [&](int k0, int dst) {
        #pragma unroll
        for (int i = 0; i < 8; ++i) {               // A: 128 rows x 64 floats
            const int idx = tid + i * 256;          // 0..2047 float4 slots
            const int row = idx >> 4;
            const int c4  = idx & 15;
            async_load_b128_to_lds(
                as_base + (uint32_t)dst * abuf + (uint32_t)(row * AS_STRIDE + (c4 << 2)) * 4u,
                hidden + (size_t)(m0 + row) * H + k0 + (c4 << 2));
        }
        #pragma unroll
        for (int i = 0; i < 4; ++i) {               // B: 64 experts x 64 floats
            const int idx = tid + i * 256;          // 0..1023 float4 slots
            const int e   = idx >> 4;
            const int c4  = idx & 15;
            async_load_b128_to_lds(
                bs_base + (uint32_t)dst * bbuf + (uint32_t)(e * BS_STRIDE + (c4 << 2)) * 4u,
                gate_w + (size_t)e * H + k0 + (c4 << 2));
        }
    };

    v8f acc0 = {}, acc1 = {}, acc2 = {}, acc3 = {};

    issue(0, 0);
    int buf = 0;
    for (int k0 = 0; k0 < H; k0 += KC) {
        const bool more = (k0 + KC) < H;
        if (more) {
            issue(k0 + KC, buf ^ 1);    // overlap next chunk's DMA with this chunk's WMMAs
            WAIT_ASYNCCNT(12);          // async loads retire in order: current chunk ready
        } else {
            WAIT_ASYNCCNT(0);
        }
        __syncthreads();

        // A frag (16x4): lanes 0-15 K={kk,kk+1}, lanes 16-31 K={kk+2,kk+3}
        const float* abase = &As[buf][(wave * 16 + n) * AS_STRIDE + (hi << 1)];
        // B frag (4x16): lane holds column n; contiguous K pair per half-wave
        const float* b0 = &Bs[buf][(n)      * BS_STRIDE + (hi << 1)];
        const float* b1 = &Bs[buf][(16 + n) * BS_STRIDE + (hi << 1)];
        const float* b2 = &Bs[buf][(32 + n) * BS_STRIDE + (hi << 1)];
        const float* b3 = &Bs[buf][(48 + n) * BS_STRIDE + (hi << 1)];

        #pragma unroll
        for (int kk = 0; kk < KC; kk += 4) {
            const v2f a = ld2(abase + kk);
            acc0 = __builtin_amdgcn_wmma_f32_16x16x4_f32(false, a, false, ld2(b0 + kk), (short)0, acc0, false, false);
            acc1 = __builtin_amdgcn_wmma_f32_16x16x4_f32(false, a, false, ld2(b1 + kk), (short)0, acc1, false, false);
            acc2 = __builtin_amdgcn_wmma_f32_16x16x4_f32(false, a, false, ld2(b2 + kk), (short)0, acc2, false, false);
            acc3 = __builtin_amdgcn_wmma_f32_16x16x4_f32(false, a, false, ld2(b3 + kk), (short)0, acc3, false, false);
        }
        __syncthreads();    // all waves done with this buffer before it is re-filled
        buf ^= 1;
    }

    // ---- spill logits (16x64 per wave) to LDS; C/D layout: VGPR v -> M=v (+8 hi half) ----
    float* lg = &As[0][0];
    {
        const int mbase = wave * 16 + (hi << 3);
        #pragma unroll
        for (int v = 0; v < 8; ++v) {
            lg[(mbase + v) * L_STRIDE + n]      = acc0[v];
            lg[(mbase + v) * L_STRIDE + 16 + n] = acc1[v];
            lg[(mbase + v) * L_STRIDE + 32 + n] = acc2[v];
            lg[(mbase + v) * L_STRIDE + 48 + n] = acc3[v];
        }
    }
    __syncthreads();

    // ---- top-2 (+bias) with first-index tie-break, then 2-way softmax ----
    if (tid < MT) {
        const float* row = &lg[tid * L_STRIDE];
        float m1 = -__builtin_inff(), m2 = -__builtin_inff();
        int   i1 = 0, i2 = 0;
        #pragma unroll 8
        for (int e = 0; e < E; ++e) {
            const float val = row[e] + bias_s[e];
            if (val > m1)      { m2 = m1; i2 = i1; m1 = val; i1 = e; }
            else if (val > m2) { m2 = val; i2 = e; }
        }
        const float e1 = __expf(m2 - m1);        // m2 <= m1, no overflow
        const float r  = 1.0f / (1.0f + e1);
        const int token = m0 + tid;
        out_w[token * 2 + 0] = r;
        out_w[token * 2 + 1] = e1 * r;
        out_i[token * 2 + 0] = i1;
        out_i[token * 2 + 1] = i2;
    }
}

extern "C" void kernel_launch(void* const* d_in, const int* in_sizes, int n_in,
                              void* d_out, int out_size, void* d_ws, size_t ws_size,
                              hipStream_t stream) {
    const float* hidden = (const float*)d_in[0];
    const float* gate_w = (const float*)d_in[1];
    const float* gate_b = (const float*)d_in[2];
    float* out_w = (float*)d_out;
    int*   out_i = (int*)d_out + (size_t)T_TOTAL * 2;   // indices follow weights
    LearnedRouter_kernel<<<T_TOTAL / MT, 256, 0, stream>>>(hidden, gate_w, gate_b, out_w, out_i);
}